// Retriever_33500744909531
// MI455X (gfx1250) — compile-verified
//
#include <hip/hip_runtime.h>

// CDNA5 / gfx1250, wave32.
typedef __attribute__((ext_vector_type(2))) float v2f;
typedef __attribute__((ext_vector_type(8))) float v8f;

#define HE_LD 144   // padded leading dim of h_e (138 real cols, rest zero)
#define HE_W  138
#define EMB   128
#define KSTEPS 36   // HE_LD / 4  (K steps of the 16x16x4 WMMA)
// packed B operand: [proj(2)][kk(36)][nt(8)][lane(32)] x v2f
#define BPACK_V2 (KSTEPS * 8 * 32)

// ---------------- h_e assembly: [ent|non_text (128)] [topic (2)] [pe rounds (8)] [pad->0]
__global__ void k_init_he(float* __restrict__ he, const float* __restrict__ ent,
                          const float* __restrict__ nont, const float* __restrict__ topic,
                          int N, int NTEXT) {
  long idx = (long)blockIdx.x * blockDim.x + threadIdx.x;
  long total = (long)N * HE_LD;
  if (idx >= total) return;
  int n = (int)(idx / HE_LD), c = (int)(idx % HE_LD);
  float v;
  if (c < EMB)          v = (n < NTEXT) ? ent[(long)n * EMB + c] : nont[c];
  else if (c < EMB + 2) v = topic[(long)n * 2 + (c - EMB)];
  else                  v = 0.0f;
  he[idx] = v;
}

__global__ void k_zero(float* __restrict__ p, long n) {
  long idx = (long)blockIdx.x * blockDim.x + threadIdx.x;
  if (idx < n) p[idx] = 0.0f;
}

// degree counts: cnt_fwd indexed by tail, cnt_rev by head
__global__ void k_count(const int* __restrict__ hid, const int* __restrict__ tid,
                        float* __restrict__ cf, float* __restrict__ cr, int T) {
  int e = blockIdx.x * blockDim.x + threadIdx.x;
  if (e >= T) return;
  atomicAdd(&cf[tid[e]], 1.0f);
  atomicAdd(&cr[hid[e]], 1.0f);
}

__global__ void k_inv(float* __restrict__ p, long n) {
  long idx = (long)blockIdx.x * blockDim.x + threadIdx.x;
  if (idx < n) p[idx] = 1.0f / fmaxf(p[idx], 1.0f);
}

// scatter-sum h_e[src,cs:cs+2] into h_e[dst,cd:cd+2]
__global__ void k_pe_scatter(float* __restrict__ he, const int* __restrict__ src,
                             const int* __restrict__ dst, int cs, int cd, int T) {
  int e = blockIdx.x * blockDim.x + threadIdx.x;
  if (e >= T) return;
  long s = (long)src[e] * HE_LD + cs;
  long d = (long)dst[e] * HE_LD + cd;
  atomicAdd(&he[d],     he[s]);
  atomicAdd(&he[d + 1], he[s + 1]);
}

__global__ void k_pe_scale(float* __restrict__ he, const float* __restrict__ inv, int cd, int N) {
  int n = blockIdx.x * blockDim.x + threadIdx.x;
  if (n >= N) return;
  float iv = inv[n];
  he[(long)n * HE_LD + cd]     *= iv;
  he[(long)n * HE_LD + cd + 1] *= iv;
}

// rel_proj[r,j] = relation_embs[r,:] @ W1[394:522, j]
__global__ void k_relproj(const float* __restrict__ rel, const float* __restrict__ W1,
                          float* __restrict__ relp, int NREL) {
  int idx = blockIdx.x * blockDim.x + threadIdx.x;
  if (idx >= NREL * EMB) return;
  int r = idx >> 7, j = idx & 127;
  const float* Wd = W1 + 394 * EMB;
  float s = 0.0f;
  #pragma unroll 8
  for (int k = 0; k < EMB; ++k) s += rel[(long)r * EMB + k] * Wd[(long)k * EMB + j];
  relp[idx] = s;
}

// base[i,j] = b1[j] + intent[i]@W1[0:128,j] + q@W1[128:256,j]
__global__ void k_base(const float* __restrict__ intents, const float* __restrict__ q,
                       const float* __restrict__ W1, const float* __restrict__ b1,
                       float* __restrict__ base, int I) {
  int idx = blockIdx.x * blockDim.x + threadIdx.x;
  if (idx >= I * EMB) return;
  int i = idx >> 7, j = idx & 127;
  float s = b1[j];
  #pragma unroll 8
  for (int k = 0; k < EMB; ++k) {
    s += intents[(long)i * EMB + k] * W1[(long)k * EMB + j];
    s += q[k] * W1[(long)(EMB + k) * EMB + j];
  }
  base[idx] = s;
}

// Pre-pack W1 head/tail slices into the exact per-lane WMMA B layout,
// zero-padded K 138->144, so the GEMM inner loop has no conditionals.
// bpack[p][kk][nt][lane] = { Wsub[ka, col], Wsub[ka+1, col] }
//   ka = kk*4 + (lane>>4)*2, col = nt*16 + (lane&15)
__global__ void k_packB(const float* __restrict__ W1, float* __restrict__ bpack) {
  int idx = blockIdx.x * blockDim.x + threadIdx.x;
  if (idx >= 2 * BPACK_V2) return;
  int lane = idx & 31;
  int nt   = (idx >> 5) & 7;
  int kk   = (idx >> 8) % KSTEPS;
  int p    = idx / BPACK_V2;
  const float* Wsub = W1 + (p ? 522 : 256) * EMB;
  int ka  = (kk << 2) + ((lane >> 4) << 1);
  int col = (nt << 4) + (lane & 15);
  float x = (ka     < HE_W) ? Wsub[(long)ka       * EMB + col] : 0.0f;
  float y = (ka + 1 < HE_W) ? Wsub[(long)(ka + 1) * EMB + col] : 0.0f;
  bpack[(long)idx * 2]     = x;
  bpack[(long)idx * 2 + 1] = y;
}

// WMMA projection: out = h_e[N,138] @ W1slice[138,128], f32 16x16x4.
// One wave computes a 16-row x 128-col stripe (8 accumulators, A reused).
// Inner loop: 1 b64 A load + 8 b64 packed-B loads + 8 WMMAs, branch-free.
__global__ void __launch_bounds__(256)
k_proj_wmma(const float* __restrict__ he, const float* __restrict__ bpack,
            float* __restrict__ headp, float* __restrict__ tailp, int N) {
  int lane  = threadIdx.x & 31;
  int gwave = blockIdx.x * (blockDim.x >> 5) + (threadIdx.x >> 5);
  int ntiles = (N + 15) >> 4;
  int tile  = gwave >> 1;
  int which = gwave & 1;          // 0 = head slice, 1 = tail slice
  if (tile >= ntiles) return;     // wave-uniform: EXEC stays all-ones for WMMA

  float* out = which ? tailp : headp;
  const v2f* Bp = (const v2f*)bpack + (size_t)which * BPACK_V2 + lane;

  int row = lane & 15;
  int kh  = lane >> 4;            // lanes 16..31 hold K+2 / M+8 halves
  int m0  = tile << 4;
  long mload = (long)((m0 + row < N) ? (m0 + row) : (N - 1));
  const v2f* Ap = (const v2f*)(he + mload * HE_LD) + kh;   // pair (ka, ka+1)

  v8f acc[8];
  #pragma unroll
  for (int nt = 0; nt < 8; ++nt)
    acc[nt] = (v8f){0.f, 0.f, 0.f, 0.f, 0.f, 0.f, 0.f, 0.f};

  for (int kk = 0; kk < KSTEPS; ++kk) {
    v2f a = Ap[kk * 2];                        // he[mload, kk*4 + kh*2 .. +1]
    const v2f* brow = Bp + kk * 256;           // 8*32 v2f per K step
    #pragma unroll
    for (int nt = 0; nt < 8; ++nt) {
      v2f b = brow[nt * 32];
      acc[nt] = __builtin_amdgcn_wmma_f32_16x16x4_f32(
          false, a, false, b, (short)0, acc[nt], false, false);
    }
  }

  #pragma unroll
  for (int nt = 0; nt < 8; ++nt) {
    #pragma unroll
    for (int i = 0; i < 8; ++i) {
      int m = m0 + i + kh * 8;                 // C/D layout: VGPR i -> rows i / i+8
      if (m < N) out[(long)m * EMB + (nt << 4) + row] = acc[nt][i];
    }
  }
}

// edge pass: one wave32 per edge; s = head_proj[h] + rel_proj[r] + tail_proj[t];
// per intent: dot(relu(s + base_i), W2) + b2
__global__ void __launch_bounds__(256)
k_edge(const int* __restrict__ hid, const int* __restrict__ rid, const int* __restrict__ tid,
       const float* __restrict__ headp, const float* __restrict__ relp,
       const float* __restrict__ tailp, const float* __restrict__ base,
       const float* __restrict__ W2, const float* __restrict__ b2,
       float* __restrict__ out, int T, int I) {
  int lane = threadIdx.x & 31;
  int e = blockIdx.x * (blockDim.x >> 5) + (threadIdx.x >> 5);
  if (e >= T) return;

  long h = hid[e], r = rid[e], t = tid[e];
  float4 hp = ((const float4*)(headp + h * EMB))[lane];
  float4 rp = ((const float4*)(relp  + r * EMB))[lane];
  float4 tp = ((const float4*)(tailp + t * EMB))[lane];
  float4 w2 = ((const float4*)W2)[lane];
  float  bb = b2[0];

  float4 s;
  s.x = hp.x + rp.x + tp.x;
  s.y = hp.y + rp.y + tp.y;
  s.z = hp.z + rp.z + tp.z;
  s.w = hp.w + rp.w + tp.w;

  for (int i = 0; i < I; ++i) {
    float4 bs = ((const float4*)(base + (long)i * EMB))[lane];
    float p = fmaxf(s.x + bs.x, 0.0f) * w2.x
            + fmaxf(s.y + bs.y, 0.0f) * w2.y
            + fmaxf(s.z + bs.z, 0.0f) * w2.z
            + fmaxf(s.w + bs.w, 0.0f) * w2.w;
    #pragma unroll
    for (int off = 16; off > 0; off >>= 1) p += __shfl_xor(p, off, 32);
    if (lane == 0) out[(long)e * I + i] = p + bb;
  }
}

extern "C" void kernel_launch(void* const* d_in, const int* in_sizes, int n_in,
                              void* d_out, int out_size, void* d_ws, size_t ws_size,
                              hipStream_t stream) {
  const int*   hid     = (const int*)d_in[0];
  const int*   rid     = (const int*)d_in[1];
  const int*   tid     = (const int*)d_in[2];
  const float* q       = (const float*)d_in[3];
  const float* ent     = (const float*)d_in[4];
  const float* rel     = (const float*)d_in[6];
  const float* topic   = (const float*)d_in[7];
  const float* nont    = (const float*)d_in[8];
  const float* intents = (const float*)d_in[9];
  const float* W1      = (const float*)d_in[10];
  const float* b1      = (const float*)d_in[11];
  const float* W2      = (const float*)d_in[12];
  const float* b2      = (const float*)d_in[13];

  int T     = in_sizes[0];
  int NTEXT = in_sizes[4] / EMB;
  int NREL  = in_sizes[6] / EMB;
  int N     = in_sizes[7] / 2;
  int I     = in_sizes[9] / EMB;

  float* ws = (float*)d_ws;
  size_t off = 0;
  float* he    = ws + off; off += (size_t)N * HE_LD;
  float* headp = ws + off; off += (size_t)N * EMB;
  float* tailp = ws + off; off += (size_t)N * EMB;
  float* relp  = ws + off; off += (size_t)NREL * EMB;
  float* base  = ws + off; off += (size_t)I * EMB;
  off = (off + 63) & ~(size_t)63;
  float* bpack = ws + off; off += (size_t)2 * BPACK_V2 * 2;
  off = (off + 63) & ~(size_t)63;
  float* cnt   = ws + off;                 // [0,N) = fwd (by tail), [N,2N) = rev (by head)

  const int B = 256;
  long heElems = (long)N * HE_LD;

  k_init_he<<<(int)((heElems + B - 1) / B), B, 0, stream>>>(he, ent, nont, topic, N, NTEXT);
  k_zero<<<(int)((2L * N + B - 1) / B), B, 0, stream>>>(cnt, 2L * N);
  k_count<<<(T + B - 1) / B, B, 0, stream>>>(hid, tid, cnt, cnt + N, T);
  k_inv<<<(int)((2L * N + B - 1) / B), B, 0, stream>>>(cnt, 2L * N);

  int gN = (N + B - 1) / B, gT = (T + B - 1) / B;
  // forward rounds: gather at head, scatter-mean into tail (inv_fwd)
  k_pe_scatter<<<gT, B, 0, stream>>>(he, hid, tid, 128, 130, T);
  k_pe_scale  <<<gN, B, 0, stream>>>(he, cnt, 130, N);
  k_pe_scatter<<<gT, B, 0, stream>>>(he, hid, tid, 130, 132, T);
  k_pe_scale  <<<gN, B, 0, stream>>>(he, cnt, 132, N);
  // reverse rounds: gather at tail, scatter-mean into head (inv_rev)
  k_pe_scatter<<<gT, B, 0, stream>>>(he, tid, hid, 128, 134, T);
  k_pe_scale  <<<gN, B, 0, stream>>>(he, cnt + N, 134, N);
  k_pe_scatter<<<gT, B, 0, stream>>>(he, tid, hid, 134, 136, T);
  k_pe_scale  <<<gN, B, 0, stream>>>(he, cnt + N, 136, N);

  k_relproj<<<(NREL * EMB + B - 1) / B, B, 0, stream>>>(rel, W1, relp, NREL);
  k_base<<<(I * EMB + B - 1) / B, B, 0, stream>>>(intents, q, W1, b1, base, I);
  k_packB<<<(2 * BPACK_V2 + B - 1) / B, B, 0, stream>>>(W1, bpack);

  int ntiles = (N + 15) / 16;
  int waves  = ntiles * 2;                  // x2: head slice + tail slice
  k_proj_wmma<<<(waves + 7) / 8, B, 0, stream>>>(he, bpack, headp, tailp, N);

  k_edge<<<(T + 7) / 8, B, 0, stream>>>(hid, rid, tid, headp, relp, tailp,
                                        base, W2, b2, (float*)d_out, T, I);
}